// VerticalGAT_39608188403994
// MI455X (gfx1250) — compile-verified
//
#include <hip/hip_runtime.h>
#include <cmath>

// Problem constants (must match reference)
#define BATCH 32
#define WW    256      // window / GAT feature size
#define DD    512      // d_model == number of GAT nodes
#define NHID  128
#define NHEADS 4
#define LRELU 0.1f
#define NEGF  (-1e9f)

typedef float v2f __attribute__((ext_vector_type(2)));
typedef float v8f __attribute__((ext_vector_type(8)));
typedef unsigned int u32;
typedef u32 u32x4 __attribute__((ext_vector_type(4)));
typedef u32 u32x8 __attribute__((ext_vector_type(8)));

__device__ __forceinline__ v8f wmma4(v2f a, v2f b, v8f c) {
  // D = A(16x4 f32) * B(4x16 f32) + C(16x16 f32)
  return __builtin_amdgcn_wmma_f32_16x16x4_f32(false, a, false, b, (short)0, c,
                                               false, false);
}

// ---------------------------------------------------------------------------
// Tensor Data Mover: async 2-D tile load Global -> LDS with row padding.
// 2-group descriptor (2-D tensor), data_size = 4 bytes, wave-level issue.
//   padIntervalCode: pad every 2<<code DWORDs;  padAmountCode: pad (code+1) DWORDs
// ---------------------------------------------------------------------------
__device__ __forceinline__ void tdm_load_2d_padded(u32 lds_byte_off, const void* gptr,
                                                   u32 tensor_d0, u32 tensor_d1,
                                                   u32 tile_d0, u32 tile_d1,
                                                   u32 stride0,
                                                   u32 padIntervalCode, u32 padAmountCode) {
  unsigned long long ga = (unsigned long long)gptr;
  u32x4 g0;
  g0[0] = 1u;                                              // count=1, user mode
  g0[1] = lds_byte_off;                                    // lds_addr
  g0[2] = (u32)ga;                                         // global_addr[31:0]
  g0[3] = (u32)((ga >> 32) & 0x01FFFFFFu) | (2u << 30);    // addr[56:32] | type=2
  u32x8 g1;
  g1[0] = (2u << 16)                                        // data_size = 4B
        | (1u << 20)                                        // pad_enable
        | (padIntervalCode << 22) | (padAmountCode << 25);
  g1[1] = (tensor_d0 & 0xFFFFu) << 16;                      // tensor_dim0[15:0]
  g1[2] = (tensor_d0 >> 16) | ((tensor_d1 & 0xFFFFu) << 16);
  g1[3] = (tensor_d1 >> 16) | (tile_d0 << 16);              // tile_dim0
  g1[4] = tile_d1;                                          // tile_dim1 (tile_dim2=0)
  g1[5] = stride0;                                          // tensor_dim0_stride[31:0]
  g1[6] = 0u;
  g1[7] = 0u;
  asm volatile("tensor_load_to_lds %0, %1" :: "s"(g0), "s"(g1) : "memory");
}

// ---------------------------------------------------------------------------
// Weight re-layout for conv: wt[o, t*512+i] = w[o, i, t]   (w: [512,512,3])
// ---------------------------------------------------------------------------
__global__ void wtrans_kernel(const float* __restrict__ w, float* __restrict__ wt) {
  int idx = blockIdx.x * 256 + threadIdx.x;
  if (idx >= 512 * 1536) return;
  int o = idx / 1536;
  int r = idx - o * 1536;
  int t = r >> 9;
  int i = r & 511;
  wt[idx] = w[o * 1536 + i * 3 + t];
}

// ---------------------------------------------------------------------------
// Conv1d (K=3, SAME) as WMMA GEMM: y[b,o,l] = sum_{t,i} wt[o,t*512+i]*x[b,l+t-1,i]+bias[o]
// x: [B,256,512]; y: [B,512,256]. Wave tile 32(o) x 64(l); 2 waves/block.
// Boundary handled branch-free: clamp position, multiply fragment by 0/1 mask.
// ---------------------------------------------------------------------------
__global__ void conv3_gemm_kernel(const float* __restrict__ x,
                                  const float* __restrict__ wt,
                                  const float* __restrict__ bias,
                                  float* __restrict__ y) {
  const int lane = threadIdx.x & 31;
  const int wave = threadIdx.x >> 5;
  const int m0 = blockIdx.x * 32;                 // o tile
  const int n0 = (blockIdx.y * 2 + wave) * 64;    // l tile
  const int z = blockIdx.z;                       // batch
  const int mr = m0 + (lane & 15);
  const int kh = (lane >> 4) * 2;
  const int nlo = lane & 15;
  const float* xb = x + (long long)z * 256 * 512;

  v8f acc[8] = {};
  for (int k = 0; k < 1536; k += 4) {
    const int ka = k + kh;
    const int t = ka >> 9;
    const int i = ka & 511;
    v2f a0 = *(const v2f*)(wt + (long long)mr * 1536 + ka);
    v2f a1 = *(const v2f*)(wt + (long long)(mr + 16) * 1536 + ka);
#pragma unroll
    for (int tt = 0; tt < 4; ++tt) {
      const int l = n0 + tt * 16 + nlo;
      const int pos = l + t - 1;
      const int posc = pos < 0 ? 0 : (pos > 255 ? 255 : pos);
      v2f b = *(const v2f*)(xb + (long long)posc * 512 + i);
      const float msk = (pos == posc) ? 1.0f : 0.0f;
      b.x *= msk;
      b.y *= msk;
      acc[tt] = wmma4(a0, b, acc[tt]);
      acc[4 + tt] = wmma4(a1, b, acc[4 + tt]);
    }
  }
  const int hi = lane >> 4;
#pragma unroll
  for (int p = 0; p < 2; ++p) {
#pragma unroll
    for (int r = 0; r < 8; ++r) {
      const int row = m0 + p * 16 + r + 8 * hi;   // output channel o
      const float bv = bias[row];
      float* dst = y + ((long long)z * 512 + row) * 256;
#pragma unroll
      for (int tt = 0; tt < 4; ++tt) dst[n0 + tt * 16 + nlo] = acc[p * 4 + tt][r] + bv;
    }
  }
}

// ---------------------------------------------------------------------------
// Generic WMMA GEMM: C[z][m,n] = act(alpha * sum_k A[m,k]*B[n,k] + bias[n])
//   A element:  A + (z/aDiv)*sAb + m*sAm + k*sAk      (AK1 => sAk==1, b64 frags)
//   B element:  B + (z%bMod)*sBb + n*sBn + k          (k-contiguous always)
//   bias[n]  :  bias + (z%bMod)*biasStride + n        (nullable)
// Wave tile 32 x 64; block = 2 waves; grid = (M/32, N/128, Z); act 0=none 1=sigmoid
// ---------------------------------------------------------------------------
template <bool AK1>
__global__ void gemm32x64_kernel(const float* __restrict__ A, long long sAm, long long sAk,
                                 long long sAb, int aDiv,
                                 const float* __restrict__ Bm, long long sBn, long long sBb,
                                 int bMod,
                                 const float* __restrict__ bias, long long biasStride,
                                 float* __restrict__ C, long long ldc, long long sCb,
                                 int K, float alpha, int act) {
  const int lane = threadIdx.x & 31;
  const int wave = threadIdx.x >> 5;
  const int m0 = blockIdx.x * 32;
  const int n0 = (blockIdx.y * 2 + wave) * 64;
  const int z = blockIdx.z;
  A += (long long)(z / aDiv) * sAb;
  Bm += (long long)(z % bMod) * sBb;
  C += (long long)z * sCb;
  const float* biasp = bias ? bias + (long long)(z % bMod) * biasStride : nullptr;

  const int mr = m0 + (lane & 15);
  const int kh = (lane >> 4) * 2;
  const int nb = n0 + (lane & 15);

  v8f acc[8] = {};
  for (int k = 0; k < K; k += 4) {
    const int ka = k + kh;
    v2f a0, a1;
    if (AK1) {
      a0 = *(const v2f*)(A + (long long)mr * sAm + ka);
      a1 = *(const v2f*)(A + (long long)(mr + 16) * sAm + ka);
    } else {
      a0.x = A[(long long)mr * sAm + (long long)ka * sAk];
      a0.y = A[(long long)mr * sAm + (long long)(ka + 1) * sAk];
      a1.x = A[(long long)(mr + 16) * sAm + (long long)ka * sAk];
      a1.y = A[(long long)(mr + 16) * sAm + (long long)(ka + 1) * sAk];
    }
    v2f b0 = *(const v2f*)(Bm + (long long)(nb +  0) * sBn + ka);
    v2f b1 = *(const v2f*)(Bm + (long long)(nb + 16) * sBn + ka);
    v2f b2 = *(const v2f*)(Bm + (long long)(nb + 32) * sBn + ka);
    v2f b3 = *(const v2f*)(Bm + (long long)(nb + 48) * sBn + ka);
    acc[0] = wmma4(a0, b0, acc[0]);
    acc[1] = wmma4(a0, b1, acc[1]);
    acc[2] = wmma4(a0, b2, acc[2]);
    acc[3] = wmma4(a0, b3, acc[3]);
    acc[4] = wmma4(a1, b0, acc[4]);
    acc[5] = wmma4(a1, b1, acc[5]);
    acc[6] = wmma4(a1, b2, acc[6]);
    acc[7] = wmma4(a1, b3, acc[7]);
  }
  const int hi = lane >> 4;
#pragma unroll
  for (int p = 0; p < 2; ++p) {
#pragma unroll
    for (int r = 0; r < 8; ++r) {
      const int row = m0 + p * 16 + r + 8 * hi;
      float* Crow = C + (long long)row * ldc;
#pragma unroll
      for (int t = 0; t < 4; ++t) {
        const int col = n0 + t * 16 + (lane & 15);
        float v = alpha * acc[p * 4 + t][r];
        if (biasp) v += biasp[col];
        if (act == 1) v = 1.0f / (1.0f + __expf(-v));
        Crow[col] = v;
      }
    }
  }
}

// ---------------------------------------------------------------------------
// e-vector kernel: e1[row] = Wh[row,:]·ai + aib ; e2[row] = Wh[row,:]·aj + ajb
// ---------------------------------------------------------------------------
__global__ void evec_kernel(const float* __restrict__ Wh, int F,
                            const float* __restrict__ ai, const float* __restrict__ aj,
                            const float* __restrict__ aib, const float* __restrict__ ajb,
                            int vecMod, float* __restrict__ e1, float* __restrict__ e2,
                            int rows) {
  const int lane = threadIdx.x & 31;
  const int wid = (blockIdx.x * blockDim.x + threadIdx.x) >> 5;
  if (wid >= rows) return;
  const int hsel = (wid >> 9) % vecMod;
  const float* wr = Wh + (long long)wid * F;
  const float* aih = ai + hsel * F;
  const float* ajh = aj + hsel * F;
  float s1 = 0.0f, s2 = 0.0f;
  for (int c = lane * 4; c < F; c += 128) {
    float4 wv = *(const float4*)(wr + c);
    float4 av = *(const float4*)(aih + c);
    float4 jv = *(const float4*)(ajh + c);
    s1 += wv.x * av.x + wv.y * av.y + wv.z * av.z + wv.w * av.w;
    s2 += wv.x * jv.x + wv.y * jv.y + wv.z * jv.z + wv.w * jv.w;
  }
#pragma unroll
  for (int off = 16; off; off >>= 1) {
    s1 += __shfl_xor(s1, off);
    s2 += __shfl_xor(s2, off);
  }
  if (lane == 0) {
    e1[wid] = s1 + aib[hsel];
    e2[wid] = s2 + ajb[hsel];
  }
}

// LDS layout for attention: Wh row-major with TDM padding: row m at smem[m*130],
// 128 payload dwords + 2 pad dwords per row (pad codes: interval 6 = 128 dw,
// amount 1 = 2 dw). e2s[512] after the 512 rows.
#define ROWP 130
#define SM_E2OFF (512 * ROWP)
#define SM_BYTES ((SM_E2OFF + 512) * 4)

__device__ __forceinline__ float gat_score(float e1v, float e2v, float adjv) {
  float t = e1v + e2v;
  float s = t > 0.0f ? t : LRELU * t;
  return (adjv > 0.0f) ? s : NEGF;
}

// ---------------------------------------------------------------------------
// Fused per-head GAT attention: softmax(mask(leakyrelu(e1+e2))) @ Wh, elu,
// store into hcat[b, n, h*128 + o]. One block per (b,h); 512 threads (16 waves),
// each wave two 16-row tiles. Wh staged to LDS by the Tensor Data Mover.
// ---------------------------------------------------------------------------
__global__ void attn_heads_kernel(const float* __restrict__ WhH,
                                  const float* __restrict__ e1h,
                                  const float* __restrict__ e2h,
                                  const float* __restrict__ adj,
                                  float* __restrict__ hcat) {
  extern __shared__ float smem[];
  float* e2s = smem + SM_E2OFF;
  const int z = blockIdx.x;        // b*4 + h
  const int b = z >> 2;
  const int h = z & 3;
  const int tid = threadIdx.x;

  const float* Wsrc = WhH + (long long)z * 512 * 128;
  if (tid < 32) {
    const u32 lds_off = (u32)(unsigned long long)(uintptr_t)smem;
    tdm_load_2d_padded(lds_off, Wsrc, /*tensor_d0=*/128, /*tensor_d1=*/512,
                       /*tile_d0=*/128, /*tile_d1=*/512, /*stride0=*/128,
                       /*padIntervalCode=*/6, /*padAmountCode=*/1);
    __builtin_amdgcn_s_wait_tensorcnt(0);
  }
  e2s[tid] = e2h[(long long)z * 512 + tid];   // 512 threads, 512 values
  __syncthreads();

  const int lane = tid & 31;
  const int wave = tid >> 5;
  const int kh = (lane >> 4) * 2;
  const int nlo = lane & 15;
  const int hi = lane >> 4;

  for (int tb = wave; tb < 32; tb += 16) {
    const int n0 = tb * 16;
    const int row = n0 + nlo;
    const float e1v = e1h[(long long)z * 512 + row];
    const float* adjr = adj + ((long long)b * 512 + row) * 512;

    float mx = -3.0e38f;
    for (int m = 0; m < 512; m += 4) {
      const int mm = m + kh;
      v2f av = *(const v2f*)(adjr + mm);
      float s0 = gat_score(e1v, e2s[mm], av.x);
      float s1 = gat_score(e1v, e2s[mm + 1], av.y);
      mx = fmaxf(mx, fmaxf(s0, s1));
    }
    mx = fmaxf(mx, __shfl_xor(mx, 16));

    float sm = 0.0f;
    for (int m = 0; m < 512; m += 4) {
      const int mm = m + kh;
      v2f av = *(const v2f*)(adjr + mm);
      sm += __expf(gat_score(e1v, e2s[mm], av.x) - mx);
      sm += __expf(gat_score(e1v, e2s[mm + 1], av.y) - mx);
    }
    sm += __shfl_xor(sm, 16);
    const float inv = 1.0f / sm;

    v8f acc[8] = {};
    for (int m = 0; m < 512; m += 4) {
      const int mm = m + kh;
      v2f av = *(const v2f*)(adjr + mm);
      v2f a;
      a.x = __expf(gat_score(e1v, e2s[mm], av.x) - mx) * inv;
      a.y = __expf(gat_score(e1v, e2s[mm + 1], av.y) - mx) * inv;
      const float* brow = smem + (long long)mm * ROWP;
#pragma unroll
      for (int j = 0; j < 8; ++j) {
        const int o = j * 16 + nlo;
        v2f bf;
        bf.x = brow[o];          // B[mm, o]
        bf.y = brow[ROWP + o];   // B[mm+1, o]
        acc[j] = wmma4(a, bf, acc[j]);
      }
    }
#pragma unroll
    for (int r = 0; r < 8; ++r) {
      const int n = n0 + r + 8 * hi;
      float* dst = hcat + ((long long)b * 512 + n) * 512 + h * 128;
#pragma unroll
      for (int j = 0; j < 8; ++j) {
        float v = acc[j][r];
        dst[j * 16 + nlo] = v > 0.0f ? v : __expf(v) - 1.0f;   // ELU
      }
    }
  }
}

// ---------------------------------------------------------------------------
// Fused output GAT attention over a 128-column half of Wh2 (512x256).
// One block per (b, half); 512 threads; result g[b, n, o0+o] (no activation).
// ---------------------------------------------------------------------------
__global__ void attn_out_kernel(const float* __restrict__ Wh2,
                                const float* __restrict__ e1o,
                                const float* __restrict__ e2o,
                                const float* __restrict__ adj,
                                float* __restrict__ g) {
  extern __shared__ float smem[];
  float* e2s = smem + SM_E2OFF;
  const int z = blockIdx.x;       // b*2 + half
  const int b = z >> 1;
  const int o0 = (z & 1) * 128;
  const int tid = threadIdx.x;

  const float* Wsrc = Wh2 + (long long)b * 512 * 256 + o0;
  if (tid < 32) {
    const u32 lds_off = (u32)(unsigned long long)(uintptr_t)smem;
    tdm_load_2d_padded(lds_off, Wsrc, /*tensor_d0=*/256, /*tensor_d1=*/512,
                       /*tile_d0=*/128, /*tile_d1=*/512, /*stride0=*/256,
                       /*padIntervalCode=*/6, /*padAmountCode=*/1);
    __builtin_amdgcn_s_wait_tensorcnt(0);
  }
  e2s[tid] = e2o[(long long)b * 512 + tid];
  __syncthreads();

  const int lane = tid & 31;
  const int wave = tid >> 5;
  const int kh = (lane >> 4) * 2;
  const int nlo = lane & 15;
  const int hi = lane >> 4;

  for (int tb = wave; tb < 32; tb += 16) {
    const int n0 = tb * 16;
    const int row = n0 + nlo;
    const float e1v = e1o[(long long)b * 512 + row];
    const float* adjr = adj + ((long long)b * 512 + row) * 512;

    float mx = -3.0e38f;
    for (int m = 0; m < 512; m += 4) {
      const int mm = m + kh;
      v2f av = *(const v2f*)(adjr + mm);
      float s0 = gat_score(e1v, e2s[mm], av.x);
      float s1 = gat_score(e1v, e2s[mm + 1], av.y);
      mx = fmaxf(mx, fmaxf(s0, s1));
    }
    mx = fmaxf(mx, __shfl_xor(mx, 16));

    float sm = 0.0f;
    for (int m = 0; m < 512; m += 4) {
      const int mm = m + kh;
      v2f av = *(const v2f*)(adjr + mm);
      sm += __expf(gat_score(e1v, e2s[mm], av.x) - mx);
      sm += __expf(gat_score(e1v, e2s[mm + 1], av.y) - mx);
    }
    sm += __shfl_xor(sm, 16);
    const float inv = 1.0f / sm;

    v8f acc[8] = {};
    for (int m = 0; m < 512; m += 4) {
      const int mm = m + kh;
      v2f av = *(const v2f*)(adjr + mm);
      v2f a;
      a.x = __expf(gat_score(e1v, e2s[mm], av.x) - mx) * inv;
      a.y = __expf(gat_score(e1v, e2s[mm + 1], av.y) - mx) * inv;
      const float* brow = smem + (long long)mm * ROWP;
#pragma unroll
      for (int j = 0; j < 8; ++j) {
        const int o = j * 16 + nlo;
        v2f bf;
        bf.x = brow[o];
        bf.y = brow[ROWP + o];
        acc[j] = wmma4(a, bf, acc[j]);
      }
    }
#pragma unroll
    for (int r = 0; r < 8; ++r) {
      const int n = n0 + r + 8 * hi;
      float* dst = g + ((long long)b * 512 + n) * 256 + o0;
#pragma unroll
      for (int j = 0; j < 8; ++j) dst[j * 16 + nlo] = acc[j][r];
    }
  }
}

// ---------------------------------------------------------------------------
// gat_y = log_softmax(elu(g)) along rows of length 256 (in place).
// ---------------------------------------------------------------------------
__global__ void elu_logsoftmax_kernel(float* __restrict__ g, int rows) {
  const int lane = threadIdx.x & 31;
  const int wid = (blockIdx.x * blockDim.x + threadIdx.x) >> 5;
  if (wid >= rows) return;
  float* row = g + (long long)wid * 256;
  float4 v0 = *(const float4*)(row + lane * 4);
  float4 v1 = *(const float4*)(row + 128 + lane * 4);
  float v[8] = {v0.x, v0.y, v0.z, v0.w, v1.x, v1.y, v1.z, v1.w};
#pragma unroll
  for (int i = 0; i < 8; ++i) v[i] = v[i] > 0.0f ? v[i] : __expf(v[i]) - 1.0f;
  float mx = v[0];
#pragma unroll
  for (int i = 1; i < 8; ++i) mx = fmaxf(mx, v[i]);
#pragma unroll
  for (int off = 16; off; off >>= 1) mx = fmaxf(mx, __shfl_xor(mx, off));
  float sm = 0.0f;
#pragma unroll
  for (int i = 0; i < 8; ++i) sm += __expf(v[i] - mx);
#pragma unroll
  for (int off = 16; off; off >>= 1) sm += __shfl_xor(sm, off);
  const float L = mx + __logf(sm);
  v0.x = v[0] - L; v0.y = v[1] - L; v0.z = v[2] - L; v0.w = v[3] - L;
  v1.x = v[4] - L; v1.y = v[5] - L; v1.z = v[6] - L; v1.w = v[7] - L;
  *(float4*)(row + lane * 4) = v0;
  *(float4*)(row + 128 + lane * 4) = v1;
}

// ---------------------------------------------------------------------------
// Host launch
// ---------------------------------------------------------------------------
extern "C" void kernel_launch(void* const* d_in, const int* in_sizes, int n_in,
                              void* d_out, int out_size, void* d_ws, size_t ws_size,
                              hipStream_t stream) {
  const float* x        = (const float*)d_in[0];
  const float* qW       = (const float*)d_in[1];
  const float* qb       = (const float*)d_in[2];
  const float* kW       = (const float*)d_in[3];
  const float* kb       = (const float*)d_in[4];
  const float* vW       = (const float*)d_in[5];
  const float* vb       = (const float*)d_in[6];
  const float* head_W   = (const float*)d_in[7];
  const float* head_b   = (const float*)d_in[8];
  const float* head_ai  = (const float*)d_in[9];
  const float* head_ai_b= (const float*)d_in[10];
  const float* head_aj  = (const float*)d_in[11];
  const float* head_aj_b= (const float*)d_in[12];
  const float* out_W    = (const float*)d_in[13];
  const float* out_b    = (const float*)d_in[14];
  const float* out_ai   = (const float*)d_in[15];
  const float* out_ai_b = (const float*)d_in[16];
  const float* out_aj   = (const float*)d_in[17];
  const float* out_aj_b = (const float*)d_in[18];
  const float* proj_W   = (const float*)d_in[19];
  const float* proj_b   = (const float*)d_in[20];

  float* outp = (float*)d_out;                        // [B, W, D]
  float* adjp = outp + (size_t)BATCH * WW * DD;       // [B, D, D]

  // Workspace layout (floats); total ~40.3M floats (~161 MB)
  float* ws = (float*)d_ws;
  const size_t WSZ = (size_t)512 * 1536;
  float* wqT  = ws;
  float* wkT  = wqT + WSZ;
  float* wvT  = wkT + WSZ;
  float* qbuf = wvT + WSZ;                              // [B,512,256]
  float* kbuf = qbuf + (size_t)BATCH * DD * WW;
  float* vbuf = kbuf + (size_t)BATCH * DD * WW;
  float* WhH  = vbuf + (size_t)BATCH * DD * WW;         // [B,H,512,128]
  float* e1h  = WhH + (size_t)BATCH * NHEADS * DD * NHID;
  float* e2h  = e1h + (size_t)BATCH * NHEADS * DD;
  float* hcat = e2h + (size_t)BATCH * NHEADS * DD;      // [B,512,512]
  float* Wh2  = hcat + (size_t)BATCH * DD * DD;         // [B,512,256]
  float* e1o  = Wh2 + (size_t)BATCH * DD * WW;
  float* e2o  = e1o + (size_t)BATCH * DD;
  float* gbuf = e2o + (size_t)BATCH * DD;               // [B,512,256]

  const float scale = 1.0f / sqrtf((float)DD);

  // 1) transpose conv weights
  wtrans_kernel<<<3072, 256, 0, stream>>>(qW, wqT);
  wtrans_kernel<<<3072, 256, 0, stream>>>(kW, wkT);
  wtrans_kernel<<<3072, 256, 0, stream>>>(vW, wvT);

  // 2) q,k,v convs (M=512, N=256, K=1536)
  conv3_gemm_kernel<<<dim3(16, 2, BATCH), 64, 0, stream>>>(x, wqT, qb, qbuf);
  conv3_gemm_kernel<<<dim3(16, 2, BATCH), 64, 0, stream>>>(x, wkT, kb, kbuf);
  conv3_gemm_kernel<<<dim3(16, 2, BATCH), 64, 0, stream>>>(x, wvT, vb, vbuf);

  // 3) adj = sigmoid(scale * q @ k^T)  (M=512, N=512, K=256) -> d_out adj slot
  gemm32x64_kernel<true><<<dim3(16, 4, BATCH), 64, 0, stream>>>(
      qbuf, 256, 1, (long long)DD * WW, 1,
      kbuf, 256, (long long)DD * WW, BATCH,
      nullptr, 0,
      adjp, 512, (long long)DD * DD, 256, scale, 1);

  // 4) Wh per head: v @ head_W[h]^T + head_b[h]  (M=512, N=128, K=256), z = b*4+h
  gemm32x64_kernel<true><<<dim3(16, 1, BATCH * NHEADS), 64, 0, stream>>>(
      vbuf, 256, 1, (long long)DD * WW, NHEADS,
      head_W, 256, (long long)NHID * WW, NHEADS,
      head_b, NHID,
      WhH, NHID, (long long)DD * NHID, 256, 1.0f, 0);

  // 5) e1/e2 per head row
  evec_kernel<<<(BATCH * NHEADS * DD) / 8, 256, 0, stream>>>(
      WhH, NHID, head_ai, head_aj, head_ai_b, head_aj_b, NHEADS,
      e1h, e2h, BATCH * NHEADS * DD);

  // 6) fused head attention -> hcat (elu applied); Wh via TDM into LDS
  attn_heads_kernel<<<BATCH * NHEADS, 512, SM_BYTES, stream>>>(WhH, e1h, e2h, adjp, hcat);

  // 7) Wh2 = hcat @ out_W^T + out_b   (M=512, N=256, K=512)
  gemm32x64_kernel<true><<<dim3(16, 2, BATCH), 64, 0, stream>>>(
      hcat, 512, 1, (long long)DD * DD, 1,
      out_W, 512, 0, 1,
      out_b, 0,
      Wh2, 256, (long long)DD * WW, 512, 1.0f, 0);

  // 8) e1/e2 for output GAT
  evec_kernel<<<(BATCH * DD) / 8, 256, 0, stream>>>(
      Wh2, WW, out_ai, out_aj, out_ai_b, out_aj_b, 1,
      e1o, e2o, BATCH * DD);

  // 9) fused output attention -> g; Wh2 half via TDM into LDS
  attn_out_kernel<<<BATCH * 2, 512, SM_BYTES, stream>>>(Wh2, e1o, e2o, adjp, gbuf);

  // 10) gat_y = log_softmax(elu(g)) in place
  elu_logsoftmax_kernel<<<(BATCH * DD) / 8, 256, 0, stream>>>(gbuf, BATCH * DD);

  // 11) out[b,w,d] = sum_e gat_y[b,e,w] * proj_W[d,e] + proj_b[d]
  gemm32x64_kernel<false><<<dim3(8, 4, BATCH), 64, 0, stream>>>(
      gbuf, 1, 256, (long long)DD * WW, 1,
      proj_W, 512, 0, 1,
      proj_b, 0,
      outp, 512, (long long)WW * DD, 512, 1.0f, 0);
}